// VolumeRenderer_27522150433310
// MI455X (gfx1250) — compile-verified
//
#include <hip/hip_runtime.h>
#include <math.h>

#define S_SAMPLES       128
#define WAVES_PER_BLOCK 8
#define RAYS_PER_WAVE   8
#define EPSILON_F       1e10f
#define ATTEN_THRESH_F  0.001f

typedef float f32x4 __attribute__((ext_vector_type(4)));  // native vector for NT stores

// Stage one radiance row (128 samples * 16B = 2048B) into LDS via the CDNA5
// async global->LDS DMA path (tracked by ASYNCcnt). Each lane moves 16B per
// instruction; the 24-bit inst offset advances BOTH the LDS and global address
// (CDNA5 ISA), so 4 instructions copy the whole row.
__device__ __forceinline__ void async_row_to_lds(unsigned lds_byte, const float* gsrc, int lane) {
  unsigned           l = lds_byte + (unsigned)(lane * 16);
  unsigned long long g = (unsigned long long)(uintptr_t)gsrc + (unsigned)(lane * 16);
  asm volatile(
      "global_load_async_to_lds_b128 %0, %1, off\n\t"
      "global_load_async_to_lds_b128 %0, %1, off offset:512\n\t"
      "global_load_async_to_lds_b128 %0, %1, off offset:1024\n\t"
      "global_load_async_to_lds_b128 %0, %1, off offset:1536"
      :
      : "v"(l), "v"(g)
      : "memory");
}

__device__ __forceinline__ float softplus_f(float x) {
  // numerically stable softplus: max(x,0) + log1p(exp(-|x|))
  return fmaxf(x, 0.0f) + log1pf(__expf(-fabsf(x)));
}

__global__ __launch_bounds__(256, 1) void nerf_volume_render(
    const float* __restrict__ rf,     // (R, S, 4) {r,g,b,sigma}
    const float* __restrict__ dvals,  // (R, S)
    const float* __restrict__ dirs,   // (R, 3)
    float* __restrict__ out,          // concat: rgb(3R), depth(R), w(RS), mask(RS), acc(R), disp(R)
    int R) {
  __shared__ float4 tile[WAVES_PER_BLOCK][2][S_SAMPLES];  // 32 KB: double buffer per wave

  const int       lane  = threadIdx.x & 31;
  const int       wv    = threadIdx.x >> 5;
  const long long gwave = (long long)blockIdx.x * WAVES_PER_BLOCK + wv;
  const long long ray0  = gwave * RAYS_PER_WAVE;
  const long long Rl    = (long long)R;
  if (ray0 >= Rl) return;

  const unsigned ldsA = (unsigned)(uintptr_t)&tile[wv][0][0];
  const unsigned ldsB = (unsigned)(uintptr_t)&tile[wv][1][0];

  const long long base_dep  = 3LL * Rl;
  const long long base_w    = 4LL * Rl;
  const long long base_m    = base_w + Rl * S_SAMPLES;
  const long long base_acc  = base_m + Rl * S_SAMPLES;
  const long long base_disp = base_acc + Rl;

  // ---- prime the pipeline: ray0 -> buffer 0
  async_row_to_lds(ldsA, rf + ray0 * (S_SAMPLES * 4), lane);

#pragma unroll 1
  for (int i = 0; i < RAYS_PER_WAVE; ++i) {
    const long long ray  = ray0 + i;
    const bool      live = (ray < Rl);

    if (i + 1 < RAYS_PER_WAVE) {
      long long nray = ray + 1;
      if (nray >= Rl) nray = Rl - 1;  // clamp: harmless re-read, never OOB
      // WAR fence: buffer (i+1)&1 was ds-read one iteration ago
      asm volatile("s_wait_dscnt 0" ::: "memory");
      async_row_to_lds((i & 1) ? ldsA : ldsB, rf + nray * (S_SAMPLES * 4), lane);
      // wait until current ray's 4 async ops (issued last iter) have landed
      asm volatile("s_wait_asynccnt 4" ::: "memory");
    } else {
      asm volatile("s_wait_asynccnt 0" ::: "memory");
    }
    if (!live) continue;

    // ---- consume staged radiance tile from LDS (ds_load_b128 x4 per lane)
    const float4* trow = (i & 1) ? &tile[wv][1][0] : &tile[wv][0][0];
    const float4  s0   = trow[4 * lane + 0];
    const float4  s1   = trow[4 * lane + 1];
    const float4  s2   = trow[4 * lane + 2];
    const float4  s3   = trow[4 * lane + 3];

    // depth row: one coalesced b128 per lane
    const float4 d   = ((const float4*)(dvals + ray * S_SAMPLES))[lane];
    const float  dnx = __shfl_down(d.x, 1);  // depth[4L+4] from lane L+1

    // wave-uniform ray direction -> scalar loads
    const int   rs  = __builtin_amdgcn_readfirstlane((int)ray);
    const float ddx = dirs[3 * rs + 0];
    const float ddy = dirs[3 * rs + 1];
    const float ddz = dirs[3 * rs + 2];
    const float dn  = sqrtf(ddx * ddx + ddy * ddy + ddz * ddz);

    const float dl0 = (d.y - d.x) * dn;
    const float dl1 = (d.z - d.y) * dn;
    const float dl2 = (d.w - d.z) * dn;
    const float dl3 = ((lane == 31) ? EPSILON_F : (dnx - d.w)) * dn;

    // e = exp(-softplus(sigma)*delta) = 1 - alpha
    const float e0 = __expf(-softplus_f(s0.w) * dl0);
    const float e1 = __expf(-softplus_f(s1.w) * dl1);
    const float e2 = __expf(-softplus_f(s2.w) * dl2);
    const float e3 = __expf(-softplus_f(s3.w) * dl3);
    const float v0 = e0 + 1e-10f;
    const float v1 = e1 + 1e-10f;
    const float v2 = e2 + 1e-10f;
    const float v3 = e3 + 1e-10f;

    // ---- exclusive cumprod over 128 samples:
    // lane-local exclusive products, then wave prefix-product (Hillis-Steele)
    const float p1 = v0;
    const float p2 = v0 * v1;
    const float p3 = p2 * v2;
    float       x  = p3 * v3;  // inclusive product of this lane's 4 samples
#pragma unroll
    for (int dd = 1; dd < 32; dd <<= 1) {
      const float t = __shfl_up(x, dd);
      if (lane >= dd) x *= t;
    }
    float excl = __shfl_up(x, 1);
    if (lane == 0) excl = 1.0f;

    const float T0 = excl;
    const float T1 = excl * p1;
    const float T2 = excl * p2;
    const float T3 = excl * p3;
    const float w0 = (1.0f - e0) * T0;
    const float w1 = (1.0f - e1) * T1;
    const float w2 = (1.0f - e2) * T2;
    const float w3 = (1.0f - e3) * T3;

    // ---- streamed write-once outputs: non-temporal coalesced 16B stores
    f32x4 wvec = {w0, w1, w2, w3};
    f32x4 mvec = {T0 > ATTEN_THRESH_F ? 1.0f : 0.0f, T1 > ATTEN_THRESH_F ? 1.0f : 0.0f,
                  T2 > ATTEN_THRESH_F ? 1.0f : 0.0f, T3 > ATTEN_THRESH_F ? 1.0f : 0.0f};
    __builtin_nontemporal_store(wvec, ((f32x4*)(out + base_w + ray * S_SAMPLES)) + lane);
    __builtin_nontemporal_store(mvec, ((f32x4*)(out + base_m + ray * S_SAMPLES)) + lane);

    // ---- per-ray reductions (5 values, butterfly across wave32)
    float rr = w0 * s0.x + w1 * s1.x + w2 * s2.x + w3 * s3.x;
    float gg = w0 * s0.y + w1 * s1.y + w2 * s2.y + w3 * s3.y;
    float bb = w0 * s0.z + w1 * s1.z + w2 * s2.z + w3 * s3.z;
    float ac = w0 + w1 + w2 + w3;
    float dp = w0 * d.x + w1 * d.y + w2 * d.z + w3 * d.w;
#pragma unroll
    for (int m = 16; m >= 1; m >>= 1) {
      rr += __shfl_xor(rr, m);
      gg += __shfl_xor(gg, m);
      bb += __shfl_xor(bb, m);
      ac += __shfl_xor(ac, m);
      dp += __shfl_xor(dp, m);
    }

    if (lane < 3) {
      const float c       = (lane == 0) ? rr : (lane == 1) ? gg : bb;
      out[3 * ray + lane] = c;
    }
    if (lane == 0) {
      out[base_dep + ray] = dp;
      out[base_acc + ray] = ac;
      const float q       = dp / ac;
      // replicate jnp.maximum NaN propagation + jnp.where(isnan, 0)
      const float disp     = (q != q) ? 0.0f : 1.0f / fmaxf(1e-10f, q);
      out[base_disp + ray] = disp;
    }
  }
}

extern "C" void kernel_launch(void* const* d_in, const int* in_sizes, int n_in,
                              void* d_out, int out_size, void* d_ws, size_t ws_size,
                              hipStream_t stream) {
  (void)n_in; (void)out_size; (void)d_ws; (void)ws_size;
  const float* rf    = (const float*)d_in[0];  // (R,S,4)
  const float* dvals = (const float*)d_in[1];  // (R,S)
  const float* dirs  = (const float*)d_in[2];  // (R,3)
  float*       out   = (float*)d_out;

  const int R      = in_sizes[1] / S_SAMPLES;
  const int waves  = (R + RAYS_PER_WAVE - 1) / RAYS_PER_WAVE;
  const int blocks = (waves + WAVES_PER_BLOCK - 1) / WAVES_PER_BLOCK;

  nerf_volume_render<<<blocks, WAVES_PER_BLOCK * 32, 0, stream>>>(rf, dvals, dirs, out, R);
}